// NormalSmoothedPQCLoss_88192858456129
// MI455X (gfx1250) — compile-verified
//
#include <hip/hip_runtime.h>
#include <math.h>

// ---------------------------------------------------------------------------
// NormalSmoothedPQCLoss for MI455X (gfx1250, wave32)
//
// Memory-bound streaming reduction: 2 x 128 MiB read @ 23.3 TB/s => ~11.5 us
// floor. Strategy: b128 non-temporal coalesced loads, *vector* register
// accumulation (pk_fma), wave-level reduction via V_WMMA_F32_16X16X4_F32
// (B = ones), deterministic two-kernel finish.
// ---------------------------------------------------------------------------

typedef float v2f __attribute__((ext_vector_type(2)));
typedef float v4f __attribute__((ext_vector_type(4)));
typedef float v8f __attribute__((ext_vector_type(8)));

static constexpr int BROWS = 16384;
static constexpr int CCOLS = 2048;
static constexpr int WAVES_PER_BLOCK = 8;            // 256 threads / wave32
static constexpr float NLOO = 2047.0f;               // C - 1

// Full 32-lane sum using one V_WMMA_F32_16X16X4_F32.
// A layout (16x4 f32): lane m in [0,16) holds A[m][0] (vgpr0), A[m][1] (vgpr1);
// lane m+16 holds A[m][2], A[m][3]. With a = {x, 0} and B = ones:
//   D[m][n] = x[m] + x[m+16]  for all n.
// C/D layout: lanes 0-15 hold rows 0..7 in their 8 VGPRs, lanes 16-31 rows
// 8..15, so sum-of-8-regs + shfl_xor(16) completes the 32-lane reduction.
__device__ __forceinline__ float wave_reduce_wmma(float x) {
  v2f a = {x, 0.0f};
  v2f b = {1.0f, 1.0f};
  v8f c = {};
  c = __builtin_amdgcn_wmma_f32_16x16x4_f32(
      /*neg_a=*/false, a, /*neg_b=*/false, b,
      /*c_mod=*/(short)0, c, /*reuse_a=*/false, /*reuse_b=*/false);
  float t = ((c[0] + c[1]) + (c[2] + c[3])) + ((c[4] + c[5]) + (c[6] + c[7]));
  t += __shfl_xor(t, 16, 32);
  return t;
}

// Kernel 1: one wave per row. Streams the row (values + targets) with NT b128
// loads, accumulates vector partials (sum / sumsq / sum(v*t)) per lane with
// packed FMAs, WMMA-reduces across the wave, then lane 0 computes the per-row
// smoothed-CDF term.
__global__ __launch_bounds__(256) void row_cdf_kernel(
    const float* __restrict__ values,
    const float* __restrict__ targets,
    float* __restrict__ row_cdf) {
  const int wave = threadIdx.x >> 5;
  const int lane = threadIdx.x & 31;
  const int row = blockIdx.x * WAVES_PER_BLOCK + wave;

  const v4f* __restrict__ vr = (const v4f*)(values + (size_t)row * CCOLS);
  const v4f* __restrict__ tr = (const v4f*)(targets + (size_t)row * CCOLS);

  v4f acc_s  = {0.0f, 0.0f, 0.0f, 0.0f};
  v4f acc_ss = {0.0f, 0.0f, 0.0f, 0.0f};
  v4f acc_st = {0.0f, 0.0f, 0.0f, 0.0f};
#pragma unroll
  for (int i = 0; i < CCOLS / (32 * 4); ++i) {       // 16 iterations
    const int c4 = i * 32 + lane;                    // coalesced float4 index
    v4f v = __builtin_nontemporal_load(vr + c4);
    v4f t = __builtin_nontemporal_load(tr + c4);
    acc_s  += v;                                     // v_pk_add_f32 x2
    acc_ss += v * v;                                 // v_pk_fma_f32 x2
    acc_st += v * t;                                 // v_pk_fma_f32 x2
  }

  // Horizontal collapse once per row, then cross-lane WMMA reduction.
  const float s  = (acc_s.x  + acc_s.y)  + (acc_s.z  + acc_s.w);
  const float ss = (acc_ss.x + acc_ss.y) + (acc_ss.z + acc_ss.w);
  const float st = (acc_st.x + acc_st.y) + (acc_st.z + acc_st.w);

  const float row_sum = wave_reduce_wmma(s);
  const float row_ssq = wave_reduce_wmma(ss);
  const float tgt     = wave_reduce_wmma(st);

  if (lane == 0) {
    const float loo_mean  = (row_sum - tgt) * (1.0f / NLOO);
    const float loo_sumsq = row_ssq - tgt * tgt;
    const float loo_var   = (loo_sumsq - NLOO * loo_mean * loo_mean)
                            * (1.0f / (NLOO - 1.0f));
    // Reference (faithfully) uses the VARIANCE as the Normal scale.
    const float z   = (tgt - loo_mean) / (loo_var * 1.41421356237309504880f);
    const float cdf = 0.5f * (1.0f + erff(z));
    row_cdf[row] = cdf;
  }
}

// Kernel 2: deterministic reduction of the 16384 per-row cdfs.
__global__ __launch_bounds__(256) void final_reduce_kernel(
    const float* __restrict__ row_cdf,
    float* __restrict__ out) {
  __shared__ float sdata[256];
  const int tid = threadIdx.x;

  float p = 0.0f;
  for (int i = tid; i < BROWS; i += 256)             // 64 coalesced adds/thread
    p += row_cdf[i];
  sdata[tid] = p;
  __syncthreads();

  if (tid < 32) {                                    // wave 0, full EXEC
    float q = 0.0f;
#pragma unroll
    for (int j = 0; j < 8; ++j)
      q += sdata[tid * 8 + j];
    const float tot = wave_reduce_wmma(q);
    if (tid == 0)
      out[0] = -tot / (float)BROWS;
  }
}

extern "C" void kernel_launch(void* const* d_in, const int* in_sizes, int n_in,
                              void* d_out, int out_size, void* d_ws, size_t ws_size,
                              hipStream_t stream) {
  const float* values  = (const float*)d_in[0];
  const float* targets = (const float*)d_in[1];
  float* row_cdf = (float*)d_ws;                      // 16384 floats = 64 KB
  float* out = (float*)d_out;

  (void)in_sizes; (void)n_in; (void)out_size; (void)ws_size;

  row_cdf_kernel<<<BROWS / WAVES_PER_BLOCK, 256, 0, stream>>>(values, targets, row_cdf);
  final_reduce_kernel<<<1, 256, 0, stream>>>(row_cdf, out);
}